// SelectiveSSM_67138928771574
// MI455X (gfx1250) — compile-verified
//
#include <hip/hip_runtime.h>

// Selective SSM scan (Mamba-style) for gfx1250 / MI455X.
//
// Roofline: ~48 MB mandatory traffic @ 23.3 TB/s => ~2.1 us floor; compute
// (~0.4 GFLOP + 64M v_exp_f32) is negligible. The recurrence is diagonal per
// (d,n) channel, so WMMA has no legitimate contraction to accelerate; the
// CDNA5 feature that matters is the async copy engine: we double-buffer
// delta/x/B/C chunks into LDS with global_load_async_to_lds_b128 and overlap
// the next chunk's fetch with the current chunk's sequential scan, fencing
// with s_wait_asynccnt.

namespace {
constexpr int kB    = 4;
constexpr int kL    = 1024;
constexpr int kDm   = 1024;
constexpr int kN    = 16;
constexpr int kT    = 64;                 // timesteps per staged chunk
constexpr int kNChunks = kL / kT;         // 16
constexpr int kDTile   = 16;              // d-values per block
constexpr int kThreads = kDTile * kN;     // 256 threads = 8 waves
// per-buffer LDS: 4 arrays (delta, x, B, C) of kT*16 floats = 16 KB
constexpr int kBufFloats = 4 * kT * 16;   // 4096 floats
}

// Async global -> LDS copy, 16 bytes per lane.
// GVS addressing: mem = SGPR_base64 + VGPR_offset32 (+0), LDS = VGPR lds_off.
__device__ __forceinline__ void async_ld_b128(unsigned lds_off,
                                              const void* gbase,
                                              unsigned gbyte_off) {
  asm volatile("global_load_async_to_lds_b128 %0, %1, %2"
               :
               : "v"(lds_off), "v"(gbyte_off), "s"(gbase)
               : "memory");
}

__global__ __launch_bounds__(kThreads)
void ssm_scan_kernel(const float* __restrict__ x,
                     const float* __restrict__ delta,
                     const float* __restrict__ Bin,
                     const float* __restrict__ Cin,
                     const float* __restrict__ A_log,
                     const float* __restrict__ Dvec,
                     float* __restrict__ out) {
  __shared__ __align__(16) float smem[2 * kBufFloats];  // 32 KB, double buffered

  const int tid = threadIdx.x;
  const int b   = blockIdx.x >> 6;            // kDm/kDTile = 64 tiles per batch
  const int d0  = (blockIdx.x & 63) << 4;
  const int n   = tid & 15;                   // state index
  const int dl  = tid >> 4;                   // local d within tile (0..15)
  const int d   = d0 + dl;

  const float* xb = x     + (size_t)b * kL * kDm;
  const float* db = delta + (size_t)b * kL * kDm;
  const float* Bb = Bin   + (size_t)b * kL * kN;
  const float* Cb = Cin   + (size_t)b * kL * kN;
  float*       ob = out   + (size_t)b * kL * kDm;

  // Low 32 bits of the flat address of a __shared__ object == LDS byte offset.
  const unsigned smem_base = (unsigned)(size_t)(&smem[0]);

  // Per-thread staging geometry (identical every chunk).
  const int r = tid >> 2;                     // row within chunk (0..63)
  const int q = tid & 3;                      // 4-float quad within 16-wide row
  const unsigned lds_me = smem_base + (unsigned)tid * 16u;

  // Per-channel constants.
  const float Aval = -__expf(A_log[d * kN + n]);
  const float Dval = Dvec[d];

  float h = 0.0f;

  // ---- stage chunk 0 into buffer 0 (4 async b128 loads per thread) ----
  {
    const unsigned gdx = (unsigned)(((0 + r) * kDm + d0 + q * 4) * 4);
    const unsigned gbc = (unsigned)((0 * kN * 4) + tid * 16);
    async_ld_b128(lds_me,            db, gdx);
    async_ld_b128(lds_me + 4096u,    xb, gdx);
    async_ld_b128(lds_me + 8192u,    Bb, gbc);
    async_ld_b128(lds_me + 12288u,   Cb, gbc);
  }

  for (int c = 0; c < kNChunks; ++c) {
    const int buf = c & 1;
    const int l0  = c * kT;

    if (c + 1 < kNChunks) {
      // Prefetch next chunk into the other buffer, then wait until only those
      // 4 loads remain outstanding (async loads complete in order per wave).
      const int l1 = l0 + kT;
      const unsigned lds_nx = lds_me + (unsigned)((buf ^ 1) * kBufFloats * 4);
      const unsigned gdx = (unsigned)(((l1 + r) * kDm + d0 + q * 4) * 4);
      const unsigned gbc = (unsigned)((l1 * kN + tid * 4) * 4);
      async_ld_b128(lds_nx,           db, gdx);
      async_ld_b128(lds_nx + 4096u,   xb, gdx);
      async_ld_b128(lds_nx + 8192u,   Bb, gbc);
      async_ld_b128(lds_nx + 12288u,  Cb, gbc);
      asm volatile("s_wait_asynccnt 0x4" ::: "memory");
    } else {
      asm volatile("s_wait_asynccnt 0x0" ::: "memory");
    }
    __syncthreads();  // staged data visible to all waves

    const float* sD = &smem[buf * kBufFloats];
    const float* sX = sD + kT * 16;
    const float* sB = sD + 2 * kT * 16;
    const float* sC = sD + 3 * kT * 16;

#pragma unroll 4
    for (int t = 0; t < kT; ++t) {
      const float dv = sD[t * 16 + dl];   // broadcast across the 16 n-lanes
      const float xv = sX[t * 16 + dl];
      const float bv = sB[t * 16 + n];    // broadcast across the 16 d-lanes
      const float cv = sC[t * 16 + n];

      const float dA = __expf(dv * Aval);            // v_exp_f32 (trans, co-exec)
      h = fmaf(dA, h, dv * bv * xv);                 // state update

      float y = cv * h;                              // 16-wide dot over n:
      y += __shfl_xor(y, 1);                         // xor masks < 16 stay
      y += __shfl_xor(y, 2);                         // inside each wave32 half
      y += __shfl_xor(y, 4);
      y += __shfl_xor(y, 8);
      if (n == 0) {
        ob[(l0 + t) * kDm + d] = fmaf(xv, Dval, y);  // y + x*D
      }
    }
    __syncthreads();  // all waves done reading buf before it is re-staged
  }
}

extern "C" void kernel_launch(void* const* d_in, const int* in_sizes, int n_in,
                              void* d_out, int out_size, void* d_ws, size_t ws_size,
                              hipStream_t stream) {
  // setup_inputs() order: x, delta, B_in, C, A_log, D  (all float32)
  const float* x     = (const float*)d_in[0];
  const float* delta = (const float*)d_in[1];
  const float* Bin   = (const float*)d_in[2];
  const float* Cin   = (const float*)d_in[3];
  const float* A_log = (const float*)d_in[4];
  const float* Dvec  = (const float*)d_in[5];
  float* out = (float*)d_out;

  const dim3 grid(kB * (kDm / kDTile));   // 4 * 64 = 256 blocks
  const dim3 block(kThreads);             // 256 threads (8 wave32s)
  ssm_scan_kernel<<<grid, block, 0, stream>>>(x, delta, Bin, Cin, A_log, Dvec, out);
}